// StructuralSimilarity_89816356094267
// MI455X (gfx1250) — compile-verified
//
#include <hip/hip_runtime.h>

typedef float v2f __attribute__((ext_vector_type(2)));
typedef float v8f __attribute__((ext_vector_type(8)));

#define WIN 11
#define IMG_H 256
#define IMG_W 256
#define OUT_H 246
#define OUT_W 246
#define BAND 16
#define IN_ROWS 26          /* BAND + WIN - 1 */
#define SA_STRIDE 257       /* odd stride -> conflict-free column walks */
#define SV_STRIDE 269       /* 256 data + 13 zero pad cols (need up to col 267) */
#define NBANDS 16           /* ceil(246/16) */

__global__ __launch_bounds__(256)
void ssim_plane_kernel(const float* __restrict__ A,
                       const float* __restrict__ B,
                       float* __restrict__ partial)
{
    __shared__ float sA[IN_ROWS * SA_STRIDE];
    __shared__ float sB[IN_ROWS * SA_STRIDE];
    __shared__ float sV[5][BAND * SV_STRIDE];
    __shared__ float wsum[8];

    const int t    = threadIdx.x;       // 0..255 (8 waves of 32)
    const int lane = t & 31;
    const int wave = t >> 5;
    const int plane = blockIdx.x;       // 0..807
    const float* pa = A + (size_t)plane * IMG_H * IMG_W;
    const float* pb = B + (size_t)plane * IMG_H * IMG_W;

    // ---- Gaussian weights (fp32, matches jnp: exp(-c^2/(2*1.5^2)) / sum) ----
    float g[WIN];
    float gsum = 0.f;
    #pragma unroll
    for (int i = 0; i < WIN; ++i) {
        float c = (float)(i - WIN / 2);
        g[i] = expf(-(c * c) / (2.f * 1.5f * 1.5f));
        gsum += g[i];
    }
    const float inv_gsum = 1.f / gsum;
    #pragma unroll
    for (int i = 0; i < WIN; ++i) g[i] *= inv_gsum;

    // ---- Precompute WMMA B operands: W[c][n] = g[c-n], c in [0,28), n = lane&15.
    // f32 B 4x16 layout: VGPR0 -> K = k0 + (lane<16 ? 0 : 2), VGPR1 -> K+1.
    v2f bm[7];
    {
        const int n    = lane & 15;
        const int koff = (lane >> 4) << 1;   // 0 or 2
        #pragma unroll
        for (int kk = 0; kk < 7; ++kk) {
            int d0 = 4 * kk + koff - n;      // g index for VGPR0
            int d1 = d0 + 1;                 // g index for VGPR1
            float w0 = 0.f, w1 = 0.f;
            if (d0 >= 0 && d0 < WIN) {
                float c = (float)(d0 - WIN / 2);
                w0 = expf(-(c * c) / (2.f * 1.5f * 1.5f)) * inv_gsum;
            }
            if (d1 >= 0 && d1 < WIN) {
                float c = (float)(d1 - WIN / 2);
                w1 = expf(-(c * c) / (2.f * 1.5f * 1.5f)) * inv_gsum;
            }
            bm[kk].x = w0;
            bm[kk].y = w1;
        }
    }

    const float c1 = 0.01f * 0.01f;
    const float c2 = 0.03f * 0.03f;

    float lsum = 0.f;

    for (int band = 0; band < NBANDS; ++band) {
        const int row0 = band * BAND;

        // ---- stage 26 input rows of a and b into LDS (coalesced) ----
        for (int i = t; i < IN_ROWS * IMG_W; i += 256) {
            int r = i >> 8;           // row within band
            int c = i & 255;          // column
            int gr = row0 + r;
            float av = 0.f, bv = 0.f;
            if (gr < IMG_H) {
                av = pa[gr * IMG_W + c];
                bv = pb[gr * IMG_W + c];
            }
            sA[r * SA_STRIDE + c] = av;
            sB[r * SA_STRIDE + c] = bv;
        }
        __syncthreads();

        // ---- vertical 11-tap blur of 5 moments; lane owns one column ----
        {
            const int col = t;
            #pragma unroll 4
            for (int r = 0; r < BAND; ++r) {
                float sa = 0.f, sb = 0.f, saa = 0.f, sbb = 0.f, sab = 0.f;
                #pragma unroll
                for (int k = 0; k < WIN; ++k) {
                    float av = sA[(r + k) * SA_STRIDE + col];
                    float bv = sB[(r + k) * SA_STRIDE + col];
                    float gk = g[k];
                    float ga = gk * av, gb = gk * bv;
                    sa  += ga;       sb  += gb;
                    saa += ga * av;  sbb += gb * bv;  sab += ga * bv;
                }
                sV[0][r * SV_STRIDE + col] = sa;
                sV[1][r * SV_STRIDE + col] = sb;
                sV[2][r * SV_STRIDE + col] = saa;
                sV[3][r * SV_STRIDE + col] = sbb;
                sV[4][r * SV_STRIDE + col] = sab;
            }
            if (t < SV_STRIDE - IMG_W) {         // zero the 13 pad columns
                #pragma unroll
                for (int q = 0; q < 5; ++q)
                    for (int r = 0; r < BAND; ++r)
                        sV[q][r * SV_STRIDE + IMG_W + t] = 0.f;
            }
        }
        __syncthreads();

        const int bh = (row0 + BAND <= OUT_H) ? BAND : (OUT_H - row0);

        // ---- horizontal blur via V_WMMA_F32_16X16X4_F32 ----
        // A 16x4 f32 layout: lane = M (mod 16); half-wave selects K pair.
        const int m    = lane & 15;
        const int koff = (lane >> 4) << 1;
        #pragma unroll
        for (int ti = 0; ti < 2; ++ti) {
            const int x0 = (wave + ti * 8) * 16;   // waves cover tiles w and w+8
            v8f res[5];
            #pragma unroll
            for (int q = 0; q < 5; ++q) {
                v8f acc = {};
                #pragma unroll
                for (int kk = 0; kk < 7; ++kk) {
                    const float* vp = &sV[q][m * SV_STRIDE + x0 + 4 * kk + koff];
                    v2f av;
                    av.x = vp[0];
                    av.y = vp[1];
                    acc = __builtin_amdgcn_wmma_f32_16x16x4_f32(
                        false, av, false, bm[kk], (short)0, acc, false, false);
                }
                res[q] = acc;
            }

            // ---- SSIM map + masked accumulation (D layout: M = r + 8*half) ----
            const int nn    = lane & 15;
            const int mbase = (lane < 16) ? 0 : 8;
            #pragma unroll
            for (int r = 0; r < 8; ++r) {
                float mu1 = res[0][r], mu2 = res[1][r];
                float xx  = res[2][r], yy  = res[3][r], xy = res[4][r];
                float m11 = mu1 * mu1, m22 = mu2 * mu2, m12 = mu1 * mu2;
                float s1  = xx - m11, s2 = yy - m22, s12 = xy - m12;
                float cs  = (2.f * s12 + c2) / (s1 + s2 + c2);
                float lum = (2.f * m12 + c1) / (m11 + m22 + c1);
                float v   = lum * cs;
                bool valid = ((mbase + r) < bh) && ((x0 + nn) < OUT_W);
                lsum += valid ? v : 0.f;
            }
        }
        __syncthreads();   // protect LDS before next band overwrites it
    }

    // ---- wave32 reduction, then cross-wave via LDS ----
    #pragma unroll
    for (int off = 16; off > 0; off >>= 1)
        lsum += __shfl_down(lsum, off, 32);
    if (lane == 0) wsum[wave] = lsum;
    __syncthreads();
    if (t == 0) {
        float s = 0.f;
        #pragma unroll
        for (int w = 0; w < 8; ++w) s += wsum[w];
        partial[plane] = s;
    }
}

__global__ __launch_bounds__(256)
void ssim_reduce_kernel(const float* __restrict__ partial,
                        float* __restrict__ out,
                        int n, float inv_count)
{
    __shared__ float sm[256];
    float s = 0.f;
    for (int i = threadIdx.x; i < n; i += 256) s += partial[i];
    sm[threadIdx.x] = s;
    __syncthreads();
    for (int off = 128; off > 0; off >>= 1) {
        if (threadIdx.x < off) sm[threadIdx.x] += sm[threadIdx.x + off];
        __syncthreads();
    }
    if (threadIdx.x == 0) out[0] = sm[0] * inv_count;
}

extern "C" void kernel_launch(void* const* d_in, const int* in_sizes, int n_in,
                              void* d_out, int out_size, void* d_ws, size_t ws_size,
                              hipStream_t stream) {
    const float* a = (const float*)d_in[0];
    const float* b = (const float*)d_in[1];
    float* out = (float*)d_out;
    float* partial = (float*)d_ws;

    const int planes = in_sizes[0] / (IMG_H * IMG_W);   // 4*202 = 808
    const double count = (double)planes * OUT_H * OUT_W;
    const float inv_count = (float)(1.0 / count);

    ssim_plane_kernel<<<planes, 256, 0, stream>>>(a, b, partial);
    ssim_reduce_kernel<<<1, 256, 0, stream>>>(partial, out, planes, inv_count);
}